// CustomSkipOperation_8521215115833
// MI455X (gfx1250) — compile-verified
//
#include <hip/hip_runtime.h>
#include <math.h>

// ---------------------------------------------------------------------------
// FBP + crop for MI455X (gfx1250, wave32).
//   x: (2,1,48,216,96) f32  ->  out: (2,1,48,96,96) f32
//
// Stage 1: ramp filtering == GEMM  X(20736x96) @ H(96x96)  via
//          V_WMMA_F32_16X16X4_F32. H generated analytically (the reference's
//          FFT ramp filter is exactly a 96-tap symmetric convolution:
//          h(0)=0.5, h(odd d)=-2/(pi d)^2, h(even)=0, scaled by pi/(2*216)).
//          B fragments are hoisted into registers and reused across 4 M-tiles
//          per wave -> inner loop is pure global_load_b64 + v_wmma.
// Stage 2: backprojection over 216 angles, filtered sinogram staged in LDS
//          (2 passes of 108 angles x 96 det = 40.5 KB), branchless mask that
//          matches the reference's clip+mask arithmetic.
// Stage 3: flip + 1.8x bilinear upsample + center-crop composed into one
//          bilinear sample of the recon volume.
// ---------------------------------------------------------------------------

#define N_ANGLES 216
#define DET      96
#define WIDTH    96
#define NSINO    96                      // 2*1*48
#define M_ROWS   (NSINO * N_ANGLES)     // 20736 rows of the filter GEMM
#define KSTEPS   (DET / 4)              // 24 WMMA K-steps
#define MT_PER_WAVE 4                   // M-tiles swept per wave (B reuse)
#define UP       172                    // int(96 * 1.8)
#define CROP0    38                     // (172 - 96) / 2
#define HALF_A   108                    // angles per LDS pass

typedef __attribute__((ext_vector_type(2))) float v2f;
typedef __attribute__((ext_vector_type(8))) float v8f;

// Time-domain ramp filter coefficient, scale pi/(2*N_ANGLES) folded in.
__device__ __forceinline__ float ramp_coef(int d) {
  const double PI = 3.14159265358979323846;
  const double SCALE = PI / (2.0 * (double)N_ANGLES);
  if (d == 0) return (float)(0.5 * SCALE);
  if (d & 1) {
    double dd = (double)d;
    return (float)(-2.0 / (PI * PI * dd * dd) * SCALE);
  }
  return 0.0f;
}

// ---------------------------------------------------------------------------
// Kernel 1: filtered = X @ H  with f32 WMMA.
// One wave (32 lanes) per (4 M-tiles x 1 N-tile). grid = (324, 6).
// A 16x4 f32 layout: lanes 0-15 -> M=0..15, VGPR0/1 = K=0/1;
//                    lanes 16-31 -> M=0..15, VGPR0/1 = K=2/3.
// B 4x16 mirrors A with N in place of M. C/D: VGPR r, lanes<16 -> M=r,
// lanes>=16 -> M=r+8, N = lane&15.
// ---------------------------------------------------------------------------
__global__ __launch_bounds__(32) void fbp_filter_wmma(
    const float* __restrict__ x, float* __restrict__ filt) {
  __shared__ float sh_h[DET];
  const int lane = threadIdx.x;
  for (int d = lane; d < DET; d += 32) sh_h[d] = ramp_coef(d);
  __syncthreads();

  const int ntile = blockIdx.y;               // 0..5
  const int ncol  = ntile * 16 + (lane & 15); // B col for this lane
  const int khalf = (lane >> 4) << 1;         // 0 or 2

  // Precompute all 24 B fragments (loop-invariant w.r.t. M): 48 VGPRs.
  v2f bfrag[KSTEPS];
#pragma unroll
  for (int kk = 0; kk < KSTEPS; ++kk) {
    const int k0 = kk * 4 + khalf;
    int d0 = k0 - ncol;
    int d1 = k0 + 1 - ncol;
    bfrag[kk].x = sh_h[d0 < 0 ? -d0 : d0];
    bfrag[kk].y = sh_h[d1 < 0 ? -d1 : d1];
  }

  const int mtile0 = blockIdx.x * MT_PER_WAVE;  // 0..1292 step 4
#pragma unroll
  for (int t = 0; t < MT_PER_WAVE; ++t) {
    const int mtile = mtile0 + t;
    const int mrow  = mtile * 16 + (lane & 15);
    const float* __restrict__ arow = x + (size_t)mrow * DET + khalf;

    v8f c = {};
#pragma unroll
    for (int kk = 0; kk < KSTEPS; ++kk) {
      v2f a = *(const v2f*)(arow + kk * 4);    // 8B-aligned (khalf even)
      // 8 args: (neg_a, A, neg_b, B, c_mod, C, reuse_a, reuse_b)
      c = __builtin_amdgcn_wmma_f32_16x16x4_f32(
          false, a, false, bfrag[kk], (short)0, c, false, false);
    }

    const int mbase = mtile * 16 + ((lane >> 4) << 3);
    float* __restrict__ orow = filt + (size_t)mbase * DET + ncol;
#pragma unroll
    for (int r = 0; r < 8; ++r) orow[(size_t)r * DET] = c[r];
  }
}

// ---------------------------------------------------------------------------
// Kernel 2: backprojection. grid = (NSINO, 6); block = 256 threads.
// Each block handles 1536 pixels of one sinogram's 96x96 recon (6 px/thread).
// Filtered sinogram staged in LDS in two 108-angle passes.
// ---------------------------------------------------------------------------
__global__ __launch_bounds__(256) void fbp_backproject(
    const float* __restrict__ filt, float* __restrict__ recon) {
  __shared__ float sh_f[HALF_A * DET];        // 41472 B
  __shared__ float sh_c[N_ANGLES];
  __shared__ float sh_s[N_ANGLES];

  const int sino = blockIdx.x;
  const int tid  = threadIdx.x;

  if (tid < N_ANGLES) {
    float ang = 3.14159265358979f * (float)tid / (float)(N_ANGLES - 1);
    float sv, cv;
    sincosf(ang, &sv, &cv);
    sh_c[tid] = cv;
    sh_s[tid] = sv;
  }

  const float* __restrict__ base = filt + (size_t)sino * N_ANGLES * DET;

  float xs[6], ys[6], acc[6];
#pragma unroll
  for (int pp = 0; pp < 6; ++pp) {
    int p = blockIdx.y * 1536 + pp * 256 + tid;      // 0..9215
    xs[pp] = (float)(p % WIDTH) - 47.5f;             // (WIDTH-1)/2
    ys[pp] = (float)(p / WIDTH) - 47.5f;
    acc[pp] = 0.0f;
  }

  for (int pass = 0; pass < 2; ++pass) {
    __syncthreads();                                  // also covers trig init
    const float4* __restrict__ src =
        (const float4*)(base + pass * HALF_A * DET);
    float4* dst = (float4*)sh_f;
    for (int i = tid; i < (HALF_A * DET) / 4; i += 256) dst[i] = src[i];
    __syncthreads();

    for (int al = 0; al < HALF_A; ++al) {
      const int a = pass * HALF_A + al;
      const float cv = sh_c[a];
      const float sv = sh_s[a];
      const float* __restrict__ row = sh_f + al * DET;
#pragma unroll
      for (int pp = 0; pp < 6; ++pp) {
        float u   = xs[pp] * cv + ys[pp] * sv + 47.5f;  // + (DET-1)/2
        float inb = (u >= 0.0f && u <= 95.0f) ? 1.0f : 0.0f;
        float uc  = fminf(fmaxf(u, 0.0f), 95.0f);
        int   i0  = (int)uc;
        int   i1  = i0 < 95 ? i0 + 1 : 95;
        float fr  = uc - (float)i0;
        acc[pp] += inb * (row[i0] * (1.0f - fr) + row[i1] * fr);
      }
    }
  }

  float* __restrict__ o = recon + (size_t)sino * (WIDTH * WIDTH)
                        + (size_t)blockIdx.y * 1536;
#pragma unroll
  for (int pp = 0; pp < 6; ++pp) o[pp * 256 + tid] = acc[pp];
}

// ---------------------------------------------------------------------------
// Kernel 3: flip + bilinear 96->172 resize + center crop 38:134, composed.
// vol[y][x] = recon[95-y][95-x]; out[i][j] = bilinear(vol, cy(i+38), cx(j+38)).
// ---------------------------------------------------------------------------
__global__ __launch_bounds__(256) void fbp_resize_crop(
    const float* __restrict__ recon, float* __restrict__ out, int total) {
  int g = blockIdx.x * 256 + threadIdx.x;
  if (g >= total) return;
  int sino = g / (WIDTH * WIDTH);
  int p    = g - sino * (WIDTH * WIDTH);
  int i    = p / WIDTH;
  int j    = p - i * WIDTH;

  const float sf = (float)WIDTH / (float)UP;
  float cy = ((float)(i + CROP0) + 0.5f) * sf - 0.5f;
  float cx = ((float)(j + CROP0) + 0.5f) * sf - 0.5f;
  cy = fminf(fmaxf(cy, 0.0f), 95.0f);
  cx = fminf(fmaxf(cx, 0.0f), 95.0f);
  int   y0 = (int)cy; int y1 = y0 < 95 ? y0 + 1 : 95; float wy = cy - (float)y0;
  int   x0 = (int)cx; int x1 = x0 < 95 ? x0 + 1 : 95; float wx = cx - (float)x0;

  const float* __restrict__ rb = recon + (size_t)sino * (WIDTH * WIDTH);
  // flipped lookup
  float v00 = rb[(95 - y0) * WIDTH + (95 - x0)];
  float v01 = rb[(95 - y0) * WIDTH + (95 - x1)];
  float v10 = rb[(95 - y1) * WIDTH + (95 - x0)];
  float v11 = rb[(95 - y1) * WIDTH + (95 - x1)];
  // axis-3 (y) first, then axis-4 (x), matching the reference order
  float r0 = v00 * (1.0f - wy) + v10 * wy;
  float r1 = v01 * (1.0f - wy) + v11 * wy;
  out[g] = r0 * (1.0f - wx) + r1 * wx;
}

// ---------------------------------------------------------------------------
extern "C" void kernel_launch(void* const* d_in, const int* in_sizes, int n_in,
                              void* d_out, int out_size, void* d_ws, size_t ws_size,
                              hipStream_t stream) {
  (void)in_sizes; (void)n_in; (void)out_size; (void)ws_size;
  const float* x   = (const float*)d_in[0];
  float* out       = (float*)d_out;
  float* filt      = (float*)d_ws;                       // 20736*96 f32 = 8 MB
  float* recon     = filt + (size_t)M_ROWS * DET;        // 96*9216 f32 = 3.5 MB

  fbp_filter_wmma<<<dim3(M_ROWS / (16 * MT_PER_WAVE), DET / 16), 32, 0, stream>>>(x, filt);
  fbp_backproject<<<dim3(NSINO, 6), 256, 0, stream>>>(filt, recon);
  const int total = NSINO * WIDTH * WIDTH;               // 884736
  fbp_resize_crop<<<(total + 255) / 256, 256, 0, stream>>>(recon, out, total);
}